// CASTLE_11948599017673
// MI455X (gfx1250) — compile-verified
//
#include <hip/hip_runtime.h>
#include <math.h>

#define EPSV 1e-12f

typedef float v2f __attribute__((ext_vector_type(2)));
typedef float v8f __attribute__((ext_vector_type(8)));

#if __has_builtin(__builtin_amdgcn_wmma_f32_16x16x4_f32)
#define HAVE_WMMA_F32 1
__device__ __forceinline__ v8f wmma4(v2f a, v2f b, v8f c) {
  // 8 args: (neg_a, A, neg_b, B, c_mod, C, reuse_a, reuse_b)
  return __builtin_amdgcn_wmma_f32_16x16x4_f32(
      false, a, false, b, (short)0, c, false, false);
}
#else
#define HAVE_WMMA_F32 0
// Shuffle-based wave-level fallback with identical semantics (D = A*B + C),
// only used if the f32 WMMA builtin is unavailable on this toolchain.
__device__ __forceinline__ v8f wmma4(v2f a, v2f b, v8f c) {
  const int L = (int)(threadIdx.x & 31);
  const int n = L & 15;
  #pragma unroll
  for (int v = 0; v < 8; ++v) {
    float d = c[v];
    const int m = v + ((L >> 4) << 3);
    #pragma unroll
    for (int k = 0; k < 4; ++k) {
      float av = __shfl(a[k & 1], m + ((k >> 1) << 4), 32);
      float bv = __shfl(b[k & 1], n + ((k >> 1) << 4), 32);
      d += av * bv;
    }
    c[v] = d;
  }
  return c;
}
#endif

// ---------------------------------------------------------------------------
// Kernel 1: masked = mask_w * (diag==EPS), W[f][i] = sqrt(sum_h masked^2)
// One block per (i,f) pair; 128 threads x float4 covers H=512 exactly.
// ---------------------------------------------------------------------------
__global__ __launch_bounds__(128) void castle_mask_kernel(
    const float* __restrict__ mask_w,
    float* __restrict__ out_masked,
    float* __restrict__ out_W) {
  const int blk = blockIdx.x;       // i*64 + f
  const int i = blk >> 6, f = blk & 63;
  const int tid = threadIdx.x;
  const float sc = (i == f) ? EPSV : 1.0f;
  const float4* src = (const float4*)(mask_w + (size_t)blk * 512);
  float4* dst = (float4*)(out_masked + (size_t)blk * 512);
  float4 v = src[tid];
  v.x *= sc; v.y *= sc; v.z *= sc; v.w *= sc;
  dst[tid] = v;
  float s = v.x * v.x + v.y * v.y + v.z * v.z + v.w * v.w;
  __shared__ float red[128];
  red[tid] = s;
  __syncthreads();
  #pragma unroll
  for (int off = 64; off > 0; off >>= 1) {
    if (tid < off) red[tid] += red[tid + off];
    __syncthreads();
  }
  if (tid == 0) out_W[f * 64 + i] = sqrtf(red[0]);  // transposed
}

// ---------------------------------------------------------------------------
// Kernel 2: fused  h0 = gelu(x@masked_i + mask_b_i)
//                  xi = h0 @ W_share + b_share
//                  out = softmax(xi @ W_out_i + b_out_i)
// Block = 256 threads (8 waves: 2 row-groups x 4 col-groups), 32 B-rows.
// ---------------------------------------------------------------------------
__global__ __launch_bounds__(256) void castle_fused_kernel(
    const float* __restrict__ x,        // (2048, 64)
    const float* __restrict__ mask_w,   // (64, 64, 512)
    const float* __restrict__ mask_b,   // (64, 512)
    const float* __restrict__ W_share,  // (512, 512)
    const float* __restrict__ b_share,  // (512,)
    const float* __restrict__ W_out,    // (64, 512, 2)
    const float* __restrict__ b_out,    // (64, 2)
    float* __restrict__ out_Out,        // (2048, 64)
    float* __restrict__ out_layer0,     // (2048, 2)
    float* __restrict__ out_sm0)        // (2048, 2)
{
  const int i  = blockIdx.x;   // input-unit index 0..63
  const int bt = blockIdx.y;   // batch tile 0..63
  const int b0 = bt * 32;
  const int tid  = (int)threadIdx.x;
  const int wave = tid >> 5;
  const int lane = tid & 31;
  const int rg  = wave >> 2;   // 0..1  (row group: 16 rows each)
  const int cg  = wave & 3;    // 0..3  (col group: 128 cols each)
  const int lh  = lane >> 4;   // 0/1
  const int l16 = lane & 15;

  __shared__ float lds_x[32 * 68];     // x tile, row stride 68 (pad)
  __shared__ float lds_h0[32 * 132];   // h0 128-col block, row stride 132
  __shared__ float lds_red[64];        // 32 rows x 2 logits

  if (tid < 64) lds_red[tid] = 0.0f;

  // Stage x tile: rows b0..b0+31 are 2048 contiguous floats in global.
  {
    const float2* xs = (const float2*)(x + (size_t)b0 * 64);
    #pragma unroll
    for (int it = 0; it < 4; ++it) {
      int idx = tid + it * 256;        // float2 index, 1024 total
      int m  = idx >> 5;               // 32 float2 per row
      int c2 = (idx & 31) << 1;
      *(float2*)&lds_x[m * 68 + c2] = xs[idx];
    }
  }
  __syncthreads();

  v8f zero = {0.f, 0.f, 0.f, 0.f, 0.f, 0.f, 0.f, 0.f};
  v8f acc[8];                          // xi tile: 16 rows x 128 cols per wave
  #pragma unroll
  for (int t = 0; t < 8; ++t) acc[t] = zero;

  const float* mwi = mask_w + (size_t)i * 64 * 512;

  for (int hb = 0; hb < 4; ++hb) {     // stream H in 128-col blocks
    const int hs = hb * 128;

    // ---- h0 block (32 x 128): x(32x64) @ masked_i(64 x 128) ----
    v8f h0a[2];
    h0a[0] = zero; h0a[1] = zero;
    #pragma unroll 4
    for (int kk = 0; kk < 16; ++kk) {
      const int k = kk * 4 + lh * 2;
      v2f a = *(const v2f*)&lds_x[(rg * 16 + l16) * 68 + k];
      const float s0 = (k     == i) ? EPSV : 1.0f;
      const float s1 = (k + 1 == i) ? EPSV : 1.0f;
      #pragma unroll
      for (int t = 0; t < 2; ++t) {
        const int n = hs + cg * 32 + t * 16 + l16;
        v2f b;
        b.x = mwi[(size_t)k * 512 + n] * s0;
        b.y = mwi[(size_t)(k + 1) * 512 + n] * s1;
        h0a[t] = wmma4(a, b, h0a[t]);
      }
    }
    // gelu(exact) + mask_b, spill block to LDS for the xi GEMM A-fragments
    #pragma unroll
    for (int t = 0; t < 2; ++t) {
      const int nl = cg * 32 + t * 16 + l16;
      const float mb = mask_b[i * 512 + hs + nl];
      #pragma unroll
      for (int v = 0; v < 8; ++v) {
        float hv = h0a[t][v] + mb;
        hv = 0.5f * hv * (1.0f + erff(hv * 0.70710678118654752f));
        lds_h0[(rg * 16 + v + lh * 8) * 132 + nl] = hv;
      }
    }
    __syncthreads();

    // ---- xi += h0blk(32x128) @ W_share[hs:hs+128, :] ----
    #pragma unroll 2
    for (int kk = 0; kk < 32; ++kk) {
      const int kl = kk * 4 + lh * 2;
      v2f a = *(const v2f*)&lds_h0[(rg * 16 + l16) * 132 + kl];
      const float* wrow = W_share + (size_t)(hs + kl) * 512;
      #pragma unroll
      for (int t = 0; t < 8; ++t) {
        const int n = cg * 128 + t * 16 + l16;
        v2f b;
        b.x = wrow[n];
        b.y = wrow[512 + n];
        acc[t] = wmma4(a, b, acc[t]);
      }
    }
    __syncthreads();
  }

  // ---- output layer: out[m][o] = sum_n (xi[m][n] + b_share[n]) * W_out[i][n][o]
  float bs[8], wox[8], woy[8];
  #pragma unroll
  for (int t = 0; t < 8; ++t) {
    const int n = cg * 128 + t * 16 + l16;
    bs[t] = b_share[n];
    float2 w = *(const float2*)&W_out[((size_t)i * 512 + n) * 2];
    wox[t] = w.x; woy[t] = w.y;
  }
  #pragma unroll
  for (int v = 0; v < 8; ++v) {
    float p0 = 0.f, p1 = 0.f;
    #pragma unroll
    for (int t = 0; t < 8; ++t) {
      float xv = acc[t][v] + bs[t];
      p0 += xv * wox[t];
      p1 += xv * woy[t];
    }
    #pragma unroll
    for (int off = 1; off < 16; off <<= 1) {  // reduce within each half-wave
      p0 += __shfl_xor(p0, off, 32);
      p1 += __shfl_xor(p1, off, 32);
    }
    if (l16 == 0) {
      const int m = rg * 16 + v + lh * 8;
      atomicAdd(&lds_red[m * 2 + 0], p0);
      atomicAdd(&lds_red[m * 2 + 1], p1);
    }
  }
  __syncthreads();

  if (tid < 32) {
    const int m = tid;
    const int b = b0 + m;
    float o0 = lds_red[m * 2 + 0] + b_out[i * 2 + 0];
    float o1 = lds_red[m * 2 + 1] + b_out[i * 2 + 1];
    float mx = fmaxf(o0, o1);
    float e0 = expf(o0 - mx), e1 = expf(o1 - mx);
    float inv = 1.0f / (e0 + e1);
    out_Out[(size_t)b * 64 + i] = e1 * inv;   // softmax last component
    if (i == 0) {
      out_layer0[b * 2 + 0] = o0;
      out_layer0[b * 2 + 1] = o1;
      out_sm0[b * 2 + 0] = e0 * inv;
      out_sm0[b * 2 + 1] = e1 * inv;
    }
  }
}

// ---------------------------------------------------------------------------
extern "C" void kernel_launch(void* const* d_in, const int* in_sizes, int n_in,
                              void* d_out, int out_size, void* d_ws, size_t ws_size,
                              hipStream_t stream) {
  (void)in_sizes; (void)n_in; (void)out_size; (void)d_ws; (void)ws_size;
  const float* x       = (const float*)d_in[0];
  const float* mask_w  = (const float*)d_in[1];
  const float* mask_b  = (const float*)d_in[2];
  const float* W_share = (const float*)d_in[3];
  const float* b_share = (const float*)d_in[4];
  const float* W_out   = (const float*)d_in[5];
  const float* b_out   = (const float*)d_in[6];

  float* out = (float*)d_out;
  // Tuple layout (flat, return order):
  float* out_W      = out;                          //  64*64      = 4096
  float* out_Out    = out + 4096;                   //  2048*64    = 131072
  float* out_l0     = out + 4096 + 131072;          //  2048*2     = 4096
  float* out_masked = out + 139264;                 //  64*64*512  = 2097152
  float* out_sm0    = out + 2236416;                //  2048*2     = 4096

  castle_mask_kernel<<<4096, 128, 0, stream>>>(mask_w, out_masked, out_W);
  castle_fused_kernel<<<dim3(64, 64), 256, 0, stream>>>(
      x, mask_w, mask_b, W_share, b_share, W_out, b_out,
      out_Out, out_l0, out_sm0);
}